// multilayerADMMsparseCodingTightFrame_36661840838925
// MI455X (gfx1250) — compile-verified
//
#include <hip/hip_runtime.h>
#include <hip/hip_bf16.h>

typedef float v2f __attribute__((ext_vector_type(2)));
typedef float v8f __attribute__((ext_vector_type(8)));

// ---------------------------------------------------------------------------
// Implicit-GEMM conv via V_WMMA_F32_16X16X4_F32 (wave32).
// Each wave owns MT 16x16 output tiles stacked along M (shared B fragment):
//   out[m][n] = alphaScale * sum_k A[m][k]*B[k][n] + betaScale * src[m][n]
//   A = im2col(in) with zero padding `pad`; B = wN (N-major, [Cout][Kpad])
// Fragment layouts per CDNA5 ISA 7.12.2:
//   A (16x4 f32):  lane l holds M=l%16, K = 2*(l>>4) + {0,1}
//   B (4x16 f32):  lane l holds N=l%16, K = 2*(l>>4) + {0,1}
//   D (16x16 f32): lane l, vgpr r holds M = r + 8*(l>>4), N = l%16
// OOB padding taps read from a zero-filled scratch row (no per-load selects);
// OOB rows/cols are never stored. EXEC stays all-ones around every WMMA.
// ---------------------------------------------------------------------------
template <int CI, int MT>
__global__ __launch_bounds__(256) void conv_wmma_kernel(
    const float* __restrict__ in,   // [NB,H,W,CI]
    const float* __restrict__ wN,   // [Cout][Kpad] N-major GEMM B matrix
    const float* __restrict__ src,  // optional epilogue addend (may be null)
    const float* __restrict__ zbuf, // >= CI zero floats (padding source)
    float* __restrict__ out,        // [NB,Ho,Wo,Cout]
    int NB, int H, int W, int Ho, int Wo, int Cout, int pad,
    float alphaScale, float betaScale, int Kpad, int nTiles, int tilesMG)
{
  const int wavesPerBlock = blockDim.x >> 5;
  const int wave = blockIdx.x * wavesPerBlock + (int)(threadIdx.x >> 5);
  if (wave >= nTiles) return;                    // wave-uniform exit
  const int lane  = threadIdx.x & 31;
  const int lmod  = lane & 15;
  const int lhalf = lane >> 4;

  const int M  = NB * Ho * Wo;
  const int mg = wave % tilesMG;
  const int nt = wave / tilesMG;
  const int m0 = mg * (16 * MT);
  const int n0 = nt << 4;

  // Per-tile A-fragment pixel coordinates for this lane
  int  bA[MT], oyA[MT], oxA[MT];
  bool mOk[MT];
  #pragma unroll
  for (int t = 0; t < MT; ++t) {
    const int mA = m0 + 16 * t + lmod;
    mOk[t] = (mA < M);
    const int mC = mOk[t] ? mA : 0;
    oxA[t] = mC % Wo;
    const int t1 = mC / Wo;
    oyA[t] = t1 % Ho;
    bA[t]  = t1 / Ho;
  }

  // B-fragment column (out-of-range columns read row 0, never stored)
  const int nB = n0 + lmod;
  const float* wrow = wN + (size_t)((nB < Cout) ? nB : 0) * Kpad;

  v8f acc[MT];
  #pragma unroll
  for (int t = 0; t < MT; ++t) acc[t] = (v8f){0.f,0.f,0.f,0.f,0.f,0.f,0.f,0.f};

  if constexpr ((CI & 3) == 0) {
    // K ordering: k = (kh*5+kw)*CI + ci -> contiguous ci runs, float2 loads
    const int koff = lhalf << 1;  // 0 or 2
    int kbase = 0;
    for (int kh = 0; kh < 5; ++kh) {
      for (int kw = 0; kw < 5; ++kw) {
        const float* ap[MT];
        #pragma unroll
        for (int t = 0; t < MT; ++t) {
          const int iy = oyA[t] + kh - pad;
          const int ix = oxA[t] + kw - pad;
          const bool ok = mOk[t] && ((unsigned)iy < (unsigned)H) &&
                          ((unsigned)ix < (unsigned)W);
          ap[t] = ok ? (in + ((size_t)(bA[t] * H + iy) * W + ix) * CI) : zbuf;
        }
        #pragma unroll 4
        for (int c4 = 0; c4 < CI; c4 += 4) {
          const v2f bf = *(const v2f*)(wrow + kbase + c4 + koff);
          #pragma unroll
          for (int t = 0; t < MT; ++t) {
            const v2f af = *(const v2f*)(ap[t] + c4 + koff);
            acc[t] = __builtin_amdgcn_wmma_f32_16x16x4_f32(
                false, af, false, bf, (short)0, acc[t], false, false);
          }
        }
        kbase += CI;
      }
    }
  } else {
    // Generic (CI==3) path: flat K loop with per-element decode, K = 25*CI
    const int K = 25 * CI;
    for (int k4 = 0; k4 < K; k4 += 4) {
      const int kk = k4 + (lhalf << 1);
      const v2f bf = *(const v2f*)(wrow + kk);  // Kpad rows zero-padded past K
      #pragma unroll
      for (int t = 0; t < MT; ++t) {
        v2f af;
        #pragma unroll
        for (int c = 0; c < 2; ++c) {
          const int k = kk + c;
          float av = 0.f;
          if (mOk[t] && k < K) {
            const int r  = k / CI;
            const int ci = k - r * CI;
            const int kh = r / 5;
            const int kw = r - kh * 5;
            const int iy = oyA[t] + kh - pad;
            const int ix = oxA[t] + kw - pad;
            if ((unsigned)iy < (unsigned)H && (unsigned)ix < (unsigned)W)
              av = in[((size_t)(bA[t] * H + iy) * W + ix) * CI + ci];
          }
          if (c == 0) af.x = av; else af.y = av;
        }
        acc[t] = __builtin_amdgcn_wmma_f32_16x16x4_f32(
            false, af, false, bf, (short)0, acc[t], false, false);
      }
    }
  }

  // Epilogue: D row m = tileBase + r + 8*lhalf, col nB
  if (nB < Cout) {
    #pragma unroll
    for (int t = 0; t < MT; ++t) {
      #pragma unroll
      for (int r = 0; r < 8; ++r) {
        const int m = m0 + 16 * t + r + (lhalf << 3);
        if (m < M) {
          const size_t o = (size_t)m * Cout + nB;
          float v = alphaScale * acc[t][r];
          if (src) v += betaScale * src[o];
          out[o] = v;
        }
      }
    }
  }
}

// ---------------------------------------------------------------------------
// Elementwise: out = a*X + b*Y + c*Z   (null pointers contribute 0)
// ---------------------------------------------------------------------------
__global__ __launch_bounds__(256) void ew_kernel(
    float* __restrict__ out, const float* __restrict__ X, float a,
    const float* __restrict__ Y, float b, const float* __restrict__ Z, float c,
    size_t n)
{
  size_t i = (size_t)blockIdx.x * blockDim.x + threadIdx.x;
  const size_t stride = (size_t)gridDim.x * blockDim.x;
  for (; i < n; i += stride) {
    float v = 0.f;
    if (X) v += a * X[i];
    if (Y) v += b * Y[i];
    if (Z) v += c * Z[i];
    out[i] = v;
  }
}

// pad x (B,128,128,3) -> (B,154,154,3), writes paddedx and y (== paddedx)
__global__ __launch_bounds__(256) void pad_kernel(
    const float* __restrict__ x, float* __restrict__ padx, float* __restrict__ y,
    int Hp, int Wp, int C, int Hx, int Wx, int off, size_t n)
{
  size_t i = (size_t)blockIdx.x * blockDim.x + threadIdx.x;
  const size_t stride = (size_t)gridDim.x * blockDim.x;
  for (; i < n; i += stride) {
    const int c  = (int)(i % C);
    size_t t = i / C;
    const int w  = (int)(t % Wp); t /= Wp;
    const int h  = (int)(t % Hp);
    const int b  = (int)(t / Hp);
    const bool inside = (h >= off) && (h < off + Hx) && (w >= off) && (w < off + Wx);
    float v = 0.f;
    if (inside) v = x[((size_t)(b * Hx + (h - off)) * Wx + (w - off)) * C + c];
    padx[i] = v;
    y[i]    = v;
  }
}

// y = mask ? (paddedx + t)/(1+rho) : t   with rho == 1
__global__ __launch_bounds__(256) void yupdate_kernel(
    const float* __restrict__ t_in, const float* __restrict__ padx,
    float* __restrict__ y, int Hp, int Wp, int C, int Hx, int Wx, int off,
    size_t n)
{
  size_t i = (size_t)blockIdx.x * blockDim.x + threadIdx.x;
  const size_t stride = (size_t)gridDim.x * blockDim.x;
  for (; i < n; i += stride) {
    size_t t = i / C;
    const int w = (int)(t % Wp); t /= Wp;
    const int h = (int)(t % Hp);
    const bool inside = (h >= off) && (h < off + Hx) && (w >= off) && (w < off + Wx);
    const float tv = t_in[i];
    y[i] = inside ? 0.5f * (padx[i] + tv) : tv;
  }
}

// 2x2 max pool: in [NB,2Hp,2Wp,C] -> out [NB,Hp,Wp,C]
__global__ __launch_bounds__(256) void maxpool_kernel(
    const float* __restrict__ in, float* __restrict__ out,
    int Hp, int Wp, int C, size_t n)
{
  size_t i = (size_t)blockIdx.x * blockDim.x + threadIdx.x;
  const size_t stride = (size_t)gridDim.x * blockDim.x;
  const int H = Hp * 2, W = Wp * 2;
  for (; i < n; i += stride) {
    const int c = (int)(i % C);
    size_t t = i / C;
    const int pw = (int)(t % Wp); t /= Wp;
    const int ph = (int)(t % Hp);
    const int b  = (int)(t / Hp);
    const size_t base = ((size_t)(b * H + 2 * ph) * W + 2 * pw) * C + c;
    const size_t rs = (size_t)W * C;
    out[i] = fmaxf(fmaxf(in[base], in[base + C]),
                   fmaxf(in[base + rs], in[base + rs + C]));
  }
}

// Unpool_LS on 2x2 blocks: amg [NB,2Hp,2Wp,C] (v), recon [NB,Hp,Wp,C] (q)
// -> z (unpooled, same shape as amg), pz = max cum-average (pooled)
__global__ __launch_bounds__(256) void unpool_ls_kernel(
    const float* __restrict__ amg, const float* __restrict__ recon,
    float* __restrict__ z, float* __restrict__ pz,
    int Hp, int Wp, int C, size_t n)
{
  size_t i = (size_t)blockIdx.x * blockDim.x + threadIdx.x;
  const size_t stride = (size_t)gridDim.x * blockDim.x;
  const int H = Hp * 2, W = Wp * 2;
  const float rdenom[4] = {0.5f, 1.f / 3.f, 0.25f, 0.2f};
  for (; i < n; i += stride) {
    const int c = (int)(i % C);
    size_t t = i / C;
    const int pw = (int)(t % Wp); t /= Wp;
    const int ph = (int)(t % Hp);
    const int b  = (int)(t / Hp);
    const size_t base = ((size_t)(b * H + 2 * ph) * W + 2 * pw) * C + c;
    const size_t rs = (size_t)W * C;
    float v[4];
    v[0] = amg[base];      v[1] = amg[base + C];
    v[2] = amg[base + rs]; v[3] = amg[base + rs + C];

    // stable descending insertion sort with index tracking (== argsort(-v))
    float s[4] = {v[0], v[1], v[2], v[3]};
    int   idx[4] = {0, 1, 2, 3};
    #pragma unroll
    for (int a_ = 1; a_ < 4; ++a_) {
      const float sv = s[a_]; const int si = idx[a_];
      int j = a_ - 1;
      while (j >= 0 && s[j] < sv) { s[j + 1] = s[j]; idx[j + 1] = idx[j]; --j; }
      s[j + 1] = sv; idx[j + 1] = si;
    }
    float cs = recon[i];
    float m = -INFINITY; int amax = 0;
    #pragma unroll
    for (int k = 0; k < 4; ++k) {
      cs += s[k];
      const float cavg = cs * rdenom[k];
      if (cavg > m) { m = cavg; amax = k; }   // strict > keeps first max
    }
    float o[4] = {v[0], v[1], v[2], v[3]};
    for (int k = 0; k <= amax; ++k) o[idx[k]] = m;  // rank <= argmax -> m
    z[base]      = o[0]; z[base + C]      = o[1];
    z[base + rs] = o[2]; z[base + rs + C] = o[3];
    pz[i] = m;
  }
}

// z = shrink(alpha - gamma, lam); gamma = gamma + z - alpha
__global__ __launch_bounds__(256) void shrink_kernel(
    const float* __restrict__ alpha, float* __restrict__ gamma,
    float* __restrict__ z, float lam, size_t n)
{
  size_t i = (size_t)blockIdx.x * blockDim.x + threadIdx.x;
  const size_t stride = (size_t)gridDim.x * blockDim.x;
  for (; i < n; i += stride) {
    const float a = alpha[i];
    const float g = gamma[i];
    const float xv = a - g;
    const float sgn = (xv > 0.f) ? 1.f : ((xv < 0.f) ? -1.f : 0.f);
    const float zv = (fabsf(xv) >= lam) ? (xv - lam * sgn) : 0.f;
    z[i] = zv;
    gamma[i] = g + zv - a;
  }
}

// Forward GEMM weights, N-major: outw[co][k] = w[k*Cout+co], zero-pad to Kpad
__global__ __launch_bounds__(256) void build_fwd_kernel(
    float* __restrict__ outw, const float* __restrict__ w,
    int K, int Cout, int Kpad, size_t n)
{
  size_t i = (size_t)blockIdx.x * blockDim.x + threadIdx.x;
  const size_t stride = (size_t)gridDim.x * blockDim.x;
  for (; i < n; i += stride) {
    const int co = (int)(i / Kpad);
    const int k  = (int)(i % Kpad);
    outw[i] = (k < K) ? w[(size_t)k * Cout + co] : 0.f;
  }
}

// Transposed-conv GEMM weights, N-major: rows = Ci (new Cout'), K' = 25*Co.
// outw[ci][(kh'*5+kw')*Co + o] = w[((4-kh')*5+(4-kw'))*Ci + ci][o]
__global__ __launch_bounds__(256) void build_tr_kernel(
    float* __restrict__ outw, const float* __restrict__ w,
    int Ci, int Co, int Kpad, size_t n)
{
  size_t i = (size_t)blockIdx.x * blockDim.x + threadIdx.x;
  const size_t stride = (size_t)gridDim.x * blockDim.x;
  const int K = 25 * Co;
  for (; i < n; i += stride) {
    const int ci = (int)(i / Kpad);
    const int kp = (int)(i % Kpad);
    float v = 0.f;
    if (kp < K) {
      const int r  = kp / Co;
      const int o  = kp - r * Co;
      const int kh = r / 5;
      const int kw = r - kh * 5;
      v = w[(size_t)(((4 - kh) * 5 + (4 - kw)) * Ci + ci) * Co + o];
    }
    outw[i] = v;
  }
}

// ---------------------------------------------------------------------------
extern "C" void kernel_launch(void* const* d_in, const int* in_sizes, int n_in,
                              void* d_out, int out_size, void* d_ws, size_t ws_size,
                              hipStream_t stream)
{
  (void)in_sizes; (void)n_in; (void)out_size; (void)ws_size;
  const float* x  = (const float*)d_in[0];  // (4,128,128,3)
  const float* w0 = (const float*)d_in[1];  // (5,5,3,96)
  const float* w1 = (const float*)d_in[2];  // (5,5,96,96)
  float* out = (float*)d_out;               // (4,71,71,96)
  float* ws  = (float*)d_ws;

  const int NB = 4;
  const size_t NPAD  = (size_t)NB * 154 * 154 * 3;
  const size_t NL0   = (size_t)NB * 150 * 150 * 96;
  const size_t NPOOL = (size_t)NB * 75 * 75 * 96;
  const size_t NL1   = (size_t)NB * 71 * 71 * 96;
  const int KP0 = 76;      // 25*3 padded to multiple of 4
  const int K1  = 2400;    // 25*96

  size_t off = 0;
  auto alloc = [&](size_t nf) {
    float* p = ws + off;
    off += (nf + 63) & ~(size_t)63;
    return p;
  };
  float* zbuf    = alloc(128);   // zero scratch for OOB padding taps
  float* paddedx = alloc(NPAD);
  float* y       = alloc(NPAD);
  float* mu0     = alloc(NPAD);
  float* tS      = alloc(NPAD);
  float* alpha0  = alloc(NL0);
  float* z0      = alloc(NL0);
  float* gamma0  = alloc(NL0);
  float* tA      = alloc(NL0);
  float* tB      = alloc(NL0);
  float* pz      = alloc(NPOOL);
  float* mu1     = alloc(NPOOL);
  float* tP      = alloc(NPOOL);
  float* alpha1  = alloc(NL1);
  float* z1      = alloc(NL1);
  float* gamma1  = alloc(NL1);
  float* tQ      = alloc(NL1);
  float* wf0 = alloc((size_t)96 * KP0);
  float* wt0 = alloc((size_t)3 * K1);
  float* wf1 = alloc((size_t)96 * K1);
  float* wt1 = alloc((size_t)96 * K1);

  auto gsz = [](size_t n) {
    size_t b = (n + 255) / 256;
    if (b > 16384) b = 16384;
    return (unsigned)b;
  };
  auto EW = [&](float* o, const float* X, float a, const float* Y, float b,
                const float* Z, float c, size_t n) {
    ew_kernel<<<gsz(n), 256, 0, stream>>>(o, X, a, Y, b, Z, c, n);
  };
  auto CONV = [&](int ci, const float* in, const float* wN, const float* src,
                  float* o, int H, int W, int Ho, int Wo, int Cout, int pad,
                  float A, float Bc, int Kpad) {
    const int M = NB * Ho * Wo;
    const int tilesM = (M + 15) / 16;
    const int tilesN = (Cout + 15) / 16;
    if (ci == 3) {
      const int tilesMG = tilesM;          // MT = 1
      const int nT = tilesMG * tilesN;
      conv_wmma_kernel<3, 1><<<(nT + 7) / 8, 256, 0, stream>>>(
          in, wN, src, zbuf, o, NB, H, W, Ho, Wo, Cout, pad, A, Bc, Kpad, nT,
          tilesMG);
    } else {
      const int tilesMG = (tilesM + 1) / 2;  // MT = 2 (shared B fragment)
      const int nT = tilesMG * tilesN;
      conv_wmma_kernel<96, 2><<<(nT + 7) / 8, 256, 0, stream>>>(
          in, wN, src, zbuf, o, NB, H, W, Ho, Wo, Cout, pad, A, Bc, Kpad, nT,
          tilesMG);
    }
  };

  // ---- zero scratch + weight prep (GEMM-B matrices, N-major, zero-padded K)
  EW(zbuf, nullptr, 0.f, nullptr, 0.f, nullptr, 0.f, 128);
  build_fwd_kernel<<<gsz((size_t)96 * KP0), 256, 0, stream>>>(wf0, w0, 75, 96, KP0, (size_t)96 * KP0);
  build_fwd_kernel<<<gsz((size_t)96 * K1), 256, 0, stream>>>(wf1, w1, K1, 96, K1, (size_t)96 * K1);
  build_tr_kernel<<<gsz((size_t)3 * K1), 256, 0, stream>>>(wt0, w0, 3, 96, K1, (size_t)3 * K1);
  build_tr_kernel<<<gsz((size_t)96 * K1), 256, 0, stream>>>(wt1, w1, 96, 96, K1, (size_t)96 * K1);

  // ---- init ----
  pad_kernel<<<gsz(NPAD), 256, 0, stream>>>(x, paddedx, y, 154, 154, 3, 128, 128, 13, NPAD);
  EW(gamma0, nullptr, 0.f, nullptr, 0.f, nullptr, 0.f, NL0);
  EW(gamma1, nullptr, 0.f, nullptr, 0.f, nullptr, 0.f, NL1);
  // a0 = 0.5*conv(y,W0); alpha0 = z0 = a0
  CONV(3, y, wf0, nullptr, alpha0, 154, 154, 150, 150, 96, 0, 0.5f, 0.f, KP0);
  EW(z0, alpha0, 1.f, nullptr, 0.f, nullptr, 0.f, NL0);
  // mu0 = y - convT(a0,W0)
  CONV(96, alpha0, wt0, y, mu0, 150, 150, 154, 154, 3, 4, -1.f, 1.f, K1);
  // pz = maxpool(z0)
  maxpool_kernel<<<gsz(NPOOL), 256, 0, stream>>>(z0, pz, 75, 75, 96, NPOOL);
  // a1 = 0.5*conv(pz,W1); alpha1 = z1 = a1
  CONV(96, pz, wf1, nullptr, alpha1, 75, 75, 71, 71, 96, 0, 0.5f, 0.f, K1);
  EW(z1, alpha1, 1.f, nullptr, 0.f, nullptr, 0.f, NL1);
  // mu1 = pz - convT(a1,W1)
  CONV(96, alpha1, wt1, pz, mu1, 71, 71, 75, 75, 96, 4, -1.f, 1.f, K1);

  // ---- ADMM iterations (NOI = 2) ----
  for (int it = 0; it < 2; ++it) {
    // temp1 = convT(alpha0,W0) - mu0 ; y = where(mask,(paddedx+temp1)/2,temp1)
    CONV(96, alpha0, wt0, mu0, tS, 150, 150, 154, 154, 3, 4, 1.f, -1.f, K1);
    yupdate_kernel<<<gsz(NPAD), 256, 0, stream>>>(tS, paddedx, y, 154, 154, 3, 128, 128, 13, NPAD);
    // dhypmu = 0.5*conv(y+mu0, W0)
    EW(tS, y, 1.f, mu0, 1.f, nullptr, 0.f, NPAD);
    CONV(3, tS, wf0, nullptr, tA, 154, 154, 150, 150, 96, 0, 0.5f, 0.f, KP0);
    // zpg0 = z0 + gamma0
    EW(tB, z0, 1.f, gamma0, 1.f, nullptr, 0.f, NL0);
    // alpha0 = zpg0 + dhypmu - 0.5*conv(convT(zpg0,W0),W0)
    CONV(96, tB, wt0, nullptr, tS, 150, 150, 154, 154, 3, 4, 1.f, 0.f, K1);
    EW(alpha0, tB, 1.f, tA, 1.f, nullptr, 0.f, NL0);
    CONV(3, tS, wf0, alpha0, alpha0, 154, 154, 150, 150, 96, 0, -0.5f, 1.f, KP0);
    // mu0 = mu0 + y - convT(alpha0,W0)
    EW(mu0, mu0, 1.f, y, 1.f, nullptr, 0.f, NPAD);
    CONV(96, alpha0, wt0, mu0, mu0, 150, 150, 154, 154, 3, 4, -1.f, 1.f, K1);

    // ---- layer 1 ----
    // amg = alpha0 - gamma0
    EW(tA, alpha0, 1.f, gamma0, -1.f, nullptr, 0.f, NL0);
    // recon = convT(alpha1,W1) - mu1
    CONV(96, alpha1, wt1, mu1, tP, 71, 71, 75, 75, 96, 4, 1.f, -1.f, K1);
    // (z0, pz) = unpool_ls(amg, recon)
    unpool_ls_kernel<<<gsz(NPOOL), 256, 0, stream>>>(tA, tP, z0, pz, 75, 75, 96, NPOOL);
    // gamma0 += z0 - alpha0
    EW(gamma0, gamma0, 1.f, z0, 1.f, alpha0, -1.f, NL0);
    // zpg1 = z1 + gamma1
    EW(tQ, z1, 1.f, gamma1, 1.f, nullptr, 0.f, NL1);
    // alpha1 = zpg1 + 0.5*conv(pz+mu1,W1) - 0.5*conv(convT(zpg1,W1),W1)
    EW(tP, pz, 1.f, mu1, 1.f, nullptr, 0.f, NPOOL);
    CONV(96, tP, wf1, tQ, alpha1, 75, 75, 71, 71, 96, 0, 0.5f, 1.f, K1);
    CONV(96, tQ, wt1, nullptr, tP, 71, 71, 75, 75, 96, 4, 1.f, 0.f, K1);
    CONV(96, tP, wf1, alpha1, alpha1, 75, 75, 71, 71, 96, 0, -0.5f, 1.f, K1);
    // mu1 = mu1 + pz - convT(alpha1,W1)
    EW(mu1, mu1, 1.f, pz, 1.f, nullptr, 0.f, NPOOL);
    CONV(96, alpha1, wt1, mu1, mu1, 71, 71, 75, 75, 96, 4, -1.f, 1.f, K1);
    // z1 = shrink(alpha1-gamma1, 0.1); gamma1 += z1 - alpha1
    shrink_kernel<<<gsz(NL1), 256, 0, stream>>>(alpha1, gamma1, z1, 0.1f, NL1);
  }

  hipMemcpyAsync(out, z1, NL1 * sizeof(float), hipMemcpyDeviceToDevice, stream);
}